// InferMasks_37495064494575
// MI455X (gfx1250) — compile-verified
//
#include <hip/hip_runtime.h>

// InferMasks: out[b,c,:] = (all_empty[b,c] & OR_{r in REL[c]} (fully_labeled[b,r] & x[b,r,:]>0)) ? 0 : EMPTY
// Memory-bound: ~1.05 GiB of traffic -> ~47us at 23.3 TB/s. No matrix structure, so no WMMA;
// the CDNA5-specific win is the async global->LDS DMA path (ASYNCcnt) for the streaming reduce,
// plus b128 vector I/O, NT cache hints, uniform scalar branching, and global_prefetch_b8.

typedef __attribute__((ext_vector_type(4))) float v4f;

#define EMPTYV (-100.0f)

static constexpr int       B_  = 8;
static constexpr int       C_  = 8;
static constexpr long long VOX = 128LL * 128LL * 128LL;  // 2,097,152 voxels per channel
static constexpr int       NCH = B_ * C_;                // 64 channels
static constexpr int       F4_PER_CH = (int)(VOX / 4);   // 524,288 float4 per channel

// s_wait_asynccnt: builtin if this toolchain declares it, else raw SOPP via inline asm.
#if defined(__has_builtin) && __has_builtin(__builtin_amdgcn_s_wait_asynccnt)
#define WAIT_ASYNCCNT(n) __builtin_amdgcn_s_wait_asynccnt(n)
#else
#define WAIT_ASYNCCNT(n) asm volatile("s_wait_asynccnt %0" ::"i"(n) : "memory")
#endif

// flags[ch] bit0 = channel contains a value != EMPTY  (all_empty == !bit0)
// flags[ch] bit1 = channel contains a value == EMPTY  (has_empty ==  bit1)

__global__ void k_init(unsigned int* __restrict__ flags) {
  if (threadIdx.x < NCH) flags[threadIdx.x] = 0u;
}

__global__ __launch_bounds__(256) void k_reduce(const float* __restrict__ x,
                                                unsigned int* __restrict__ flags) {
  constexpr int SLICES     = 64;                   // blocks per channel
  constexpr int F4_PER_BLK = F4_PER_CH / SLICES;   // 8192 float4 per block
  constexpr int CHUNK      = 256;                  // float4 per chunk (1 per thread)
  constexpr int NCHUNK     = F4_PER_BLK / CHUNK;   // 32 chunks per block

  const int ch = blockIdx.x / SLICES;
  const int sl = blockIdx.x - ch * SLICES;

  // Uniform per-block SGPR base (channel start); per-lane 32-bit byte offset < 8 MiB.
  const float* __restrict__ chbase = x + (long long)ch * VOX;
  const unsigned laneF4base = (unsigned)(sl * F4_PER_BLK) + threadIdx.x;

  // Double-buffered staging: each lane owns one 16B slot per buffer.
  __shared__ v4f smem[2][CHUNK];
  // Flat shared-memory addresses truncate to the LDS byte offset (addr[31:0]).
  const unsigned ldsSlot0 =
      (unsigned)(size_t)(&smem[0][threadIdx.x]);
  const unsigned ldsSlot1 =
      (unsigned)(size_t)(&smem[1][threadIdx.x]);

  // stage(k): async DMA 16B/lane from global to this lane's LDS slot (ASYNCcnt++)
  auto stage = [&](int k) {
    const unsigned voff = (laneF4base + (unsigned)(k * CHUNK)) * 16u;  // byte offset
    const unsigned dst  = (k & 1) ? ldsSlot1 : ldsSlot0;
    asm volatile("global_load_async_to_lds_b128 %0, %1, %2"
                 :
                 : "v"(dst), "v"(voff), "s"(chbase)
                 : "memory");
  };

  stage(0);
  unsigned int local = 0u;
  for (int k = 0; k < NCHUNK; ++k) {
    if (k + 1 < NCHUNK) {
      stage(k + 1);
      WAIT_ASYNCCNT(1);   // async loads complete in order -> chunk k is in LDS
    } else {
      WAIT_ASYNCCNT(0);   // drain
    }
    v4f v = smem[k & 1][threadIdx.x];   // ds_load_b128 from own slot (no barrier needed)
    const bool anyNon =
        (v.x != EMPTYV) | (v.y != EMPTYV) | (v.z != EMPTYV) | (v.w != EMPTYV);
    const bool anyEmp =
        (v.x == EMPTYV) | (v.y == EMPTYV) | (v.z == EMPTYV) | (v.w == EMPTYV);
    local |= (anyNon ? 1u : 0u) | (anyEmp ? 2u : 0u);
    if (local == 3u) break;   // OR is monotonic; outstanding DMAs only touch our LDS slots
  }

  __shared__ unsigned int sflag;
  if (threadIdx.x == 0) sflag = 0u;
  __syncthreads();
  if (local) atomicOr(&sflag, local);
  __syncthreads();
  if (threadIdx.x == 0 && sflag) atomicOr(&flags[ch], sflag);
}

__device__ __constant__ int REL[8][2] = {
    {1, 2}, {0, 3}, {4, 5}, {6, 7}, {0, 1}, {2, 3}, {4, 5}, {6, 0}};

__global__ __launch_bounds__(256) void k_apply(const float* __restrict__ x,
                                               const unsigned int* __restrict__ flags,
                                               float* __restrict__ out) {
  constexpr int TILES       = 256;                 // blocks per channel
  constexpr int F4_PER_TILE = F4_PER_CH / TILES;   // 2048 float4 per block
  const int chIdx = blockIdx.x / TILES;
  const int tile  = blockIdx.x - chIdx * TILES;
  const int b     = chIdx / C_;
  const int c     = chIdx - b * C_;
  const long long tileOff = (long long)tile * F4_PER_TILE;

  v4f* __restrict__ o = (v4f*)(out + (long long)chIdx * VOX) + tileOff;
  const v4f E4 = {EMPTYV, EMPTYV, EMPTYV, EMPTYV};

  // Flags are uniform across the block -> scalar branches, no per-lane divergence.
  const bool all_empty = (flags[chIdx] & 1u) == 0u;
  const int  r0 = REL[c][0];
  const int  r1 = REL[c][1];
  const bool v0 = all_empty && ((flags[b * C_ + r0] & 2u) == 0u);
  const bool v1 = all_empty && ((flags[b * C_ + r1] & 2u) == 0u);

  if (!v0 && !v1) {
    // Hot path (most channels): pure EMPTY fill, zero loads, NT b128 stores.
    for (int i = threadIdx.x; i < F4_PER_TILE; i += 256)
      __builtin_nontemporal_store(E4, &o[i]);
    return;
  }

  const v4f* __restrict__ p0 =
      (const v4f*)(x + (long long)(b * C_ + r0) * VOX) + tileOff;
  const v4f* __restrict__ p1 =
      (const v4f*)(x + (long long)(b * C_ + r1) * VOX) + tileOff;

  if (v0 && v1) {
    for (int i = threadIdx.x; i < F4_PER_TILE; i += 256) {
      if (i + 256 < F4_PER_TILE) {
        __builtin_prefetch(&p0[i + 256], 0, 0);    // -> global_prefetch_b8
        __builtin_prefetch(&p1[i + 256], 0, 0);
      }
      v4f a  = p0[i];
      v4f b4 = p1[i];
      v4f r;
      r.x = (a.x > 0.0f || b4.x > 0.0f) ? 0.0f : EMPTYV;
      r.y = (a.y > 0.0f || b4.y > 0.0f) ? 0.0f : EMPTYV;
      r.z = (a.z > 0.0f || b4.z > 0.0f) ? 0.0f : EMPTYV;
      r.w = (a.w > 0.0f || b4.w > 0.0f) ? 0.0f : EMPTYV;
      __builtin_nontemporal_store(r, &o[i]);
    }
  } else {
    const v4f* __restrict__ p = v0 ? p0 : p1;
    for (int i = threadIdx.x; i < F4_PER_TILE; i += 256) {
      if (i + 256 < F4_PER_TILE) __builtin_prefetch(&p[i + 256], 0, 0);
      v4f a = p[i];
      v4f r;
      r.x = (a.x > 0.0f) ? 0.0f : EMPTYV;
      r.y = (a.y > 0.0f) ? 0.0f : EMPTYV;
      r.z = (a.z > 0.0f) ? 0.0f : EMPTYV;
      r.w = (a.w > 0.0f) ? 0.0f : EMPTYV;
      __builtin_nontemporal_store(r, &o[i]);
    }
  }
}

extern "C" void kernel_launch(void* const* d_in, const int* in_sizes, int n_in,
                              void* d_out, int out_size, void* d_ws, size_t ws_size,
                              hipStream_t stream) {
  const float*  x     = (const float*)d_in[0];
  unsigned int* flags = (unsigned int*)d_ws;   // 64 * 4 = 256 bytes of scratch
  float*        out   = (float*)d_out;

  (void)in_sizes; (void)n_in; (void)out_size; (void)ws_size;

  // Stream-ordered three-phase pipeline (graph-capture safe, deterministic):
  k_init  <<<1,          64,  0, stream>>>(flags);
  k_reduce<<<NCH * 64,   256, 0, stream>>>(x, flags);
  k_apply <<<NCH * 256,  256, 0, stream>>>(x, flags, out);
}